// BernsteinBasis_87900800680585
// MI455X (gfx1250) — compile-verified
//
#include <hip/hip_runtime.h>
#include <hip/hip_bf16.h>

// BernsteinBasis graph filter for MI455X (gfx1250, wave32).
//
// Pipeline (all in-place in d_out, which is (K+1=11, N, C) f32):
//   plane 0 <- x                                   (async d2d copy)
//   for m = 1..10:
//     plane m <- 0.5 * plane (m-1)                 (halve_kernel)
//     plane m += -0.5 * edge_val * gather(plane m-1)[col] scattered to row
//                                                  (spmm_scatter_kernel, f32 atomics)
//   out <- coeffs(11x11) @ planes                  (combine_wmma_kernel, V_WMMA_F32_16X16X4_F32)

typedef __attribute__((ext_vector_type(2))) float v2f;
typedef __attribute__((ext_vector_type(8))) float v8f;

#define KORD 10   // Bernstein order; K+1 = 11 basis functions / planes

// Bernstein coefficient matrix entry M[k,m] = (-1)^(m-k) C(10,k) C(10-k,m-k), m>=k.
// Max magnitude 4200 -> exact in f32. Columns/rows > 10 are zero padding.
__device__ __forceinline__ float bernc(int k, int m) {
    if (k > KORD || m > KORD || m < k) return 0.0f;
    int j = m - k;
    float c1 = 1.0f;
    for (int i = 0; i < k; ++i) c1 = c1 * (float)(KORD - i) / (float)(i + 1);
    float c2 = 1.0f;
    for (int i = 0; i < j; ++i) c2 = c2 * (float)(KORD - k - i) / (float)(i + 1);
    return ((j & 1) ? -c1 : c1) * c2;
}

// plane_out[i] = 0.5f * plane_in[i], float4-vectorized.
__global__ void halve_kernel(const float4* __restrict__ in, float4* __restrict__ out, int n4) {
    int i = blockIdx.x * blockDim.x + threadIdx.x;
    if (i < n4) {
        float4 v = in[i];
        v.x *= 0.5f; v.y *= 0.5f; v.z *= 0.5f; v.w *= 0.5f;
        out[i] = v;
    }
}

// 12 threads per edge, each thread handles a float4 channel block (C=48).
// dst[row] += -0.5 * w * src[col]  via native f32 global atomics (relaxed/agent).
__global__ void spmm_scatter_kernel(const float* __restrict__ src,
                                    float* __restrict__ dst,
                                    const int* __restrict__ erow,
                                    const int* __restrict__ ecol,
                                    const float* __restrict__ eval,
                                    int etot) {
    int t = blockIdx.x * blockDim.x + threadIdx.x;
    if (t >= etot * 12) return;
    int e  = t / 12;
    int cb = (t - e * 12) * 4;        // channel block start (0,4,...,44)
    int row = erow[e];
    int col = ecol[e];
    float w = -0.5f * eval[e];
    const float4 v = *reinterpret_cast<const float4*>(src + col * 48 + cb);
    float* d = dst + row * 48 + cb;
    __hip_atomic_fetch_add(d + 0, w * v.x, __ATOMIC_RELAXED, __HIP_MEMORY_SCOPE_AGENT);
    __hip_atomic_fetch_add(d + 1, w * v.y, __ATOMIC_RELAXED, __HIP_MEMORY_SCOPE_AGENT);
    __hip_atomic_fetch_add(d + 2, w * v.z, __ATOMIC_RELAXED, __HIP_MEMORY_SCOPE_AGENT);
    __hip_atomic_fetch_add(d + 3, w * v.w, __ATOMIC_RELAXED, __HIP_MEMORY_SCOPE_AGENT);
}

// In-place combine: out[k, j] = sum_{m=0..10} coeff[k,m] * planes[m, j].
// Two adjacent 16-column tiles per loop step -> two independent WMMA chains
// (hides the WMMA->WMMA accumulator RAW hazard), 6x V_WMMA_F32_16X16X4_F32.
//
// Fragment layouts per CDNA5 ISA (wave32):
//   A (16x4, f32, 2 VGPRs): M = lane&15; lanes 0-15 hold K={0,1}, lanes 16-31 K={2,3}.
//   B (4x16, f32, 2 VGPRs): N = lane&15; lanes 0-15 rows {0,1}, lanes 16-31 rows {2,3}.
//   C/D (16x16, 8 VGPRs):   VGPR r: lanes 0-15 -> M=r, lanes 16-31 -> M=r+8.
//
// All offsets fit in int (11*4.8M floats = 211 MB < 2^31 bytes).
__global__ void combine_wmma_kernel(float* __restrict__ planes,
                                    int NC,        // plane stride = N*C
                                    int npairs) {  // (N*C)/32
    const int lane = threadIdx.x & 31;
    const int colL = lane & 15;
    const int half = lane >> 4;       // 0: lanes 0-15, 1: lanes 16-31

    // Loop-invariant A fragments (coefficient matrix, rows = output plane k).
    v2f a0, a1, a2;
    a0.x = bernc(colL, 0 + 2 * half);  a0.y = bernc(colL, 1 + 2 * half);
    a1.x = bernc(colL, 4 + 2 * half);  a1.y = bernc(colL, 5 + 2 * half);
    a2.x = bernc(colL, 8 + 2 * half);  a2.y = bernc(colL, 9 + 2 * half);

    // Per-lane load offsets (plane row * NC + lane column). Row 11 is dead
    // (A column 11 is all-zero), so clamp it to plane 10: unconditional load.
    const int lo0x = (0 + 2 * half) * NC + colL;
    const int lo0y = (1 + 2 * half) * NC + colL;
    const int lo1x = (4 + 2 * half) * NC + colL;
    const int lo1y = (5 + 2 * half) * NC + colL;
    const int lo2x = (8 + 2 * half) * NC + colL;
    const int lo2y = min(9 + 2 * half, KORD) * NC + colL;

    // Per-lane store offsets: D row r -> plane r + 8*half.
    int so[8];
    #pragma unroll
    for (int r = 0; r < 8; ++r) so[r] = (r + 8 * half) * NC + colL;

    const int wave   = blockIdx.x * (blockDim.x >> 5) + (threadIdx.x >> 5);
    const int nwaves = gridDim.x * (blockDim.x >> 5);

    for (int p = wave; p < npairs; p += nwaves) {  // uniform per wave -> EXEC full
        const int j0 = p * 32;       // tile A columns j0..j0+15
        const int j1 = j0 + 16;      // tile B columns j1..j1+15

        v2f b0a, b1a, b2a, b0b, b1b, b2b;
        b0a.x = planes[lo0x + j0];  b0b.x = planes[lo0x + j1];
        b0a.y = planes[lo0y + j0];  b0b.y = planes[lo0y + j1];
        b1a.x = planes[lo1x + j0];  b1b.x = planes[lo1x + j1];
        b1a.y = planes[lo1y + j0];  b1b.y = planes[lo1y + j1];
        b2a.x = planes[lo2x + j0];  b2b.x = planes[lo2x + j1];
        b2a.y = planes[lo2y + j0];  b2b.y = planes[lo2y + j1];

        v8f ca = {};
        v8f cb = {};
        // Two independent accumulator chains, interleaved.
        ca = __builtin_amdgcn_wmma_f32_16x16x4_f32(false, a0, false, b0a, (short)0, ca, false, false);
        cb = __builtin_amdgcn_wmma_f32_16x16x4_f32(false, a0, false, b0b, (short)0, cb, false, false);
        ca = __builtin_amdgcn_wmma_f32_16x16x4_f32(false, a1, false, b1a, (short)0, ca, false, false);
        cb = __builtin_amdgcn_wmma_f32_16x16x4_f32(false, a1, false, b1b, (short)0, cb, false, false);
        ca = __builtin_amdgcn_wmma_f32_16x16x4_f32(false, a2, false, b2a, (short)0, ca, false, false);
        cb = __builtin_amdgcn_wmma_f32_16x16x4_f32(false, a2, false, b2b, (short)0, cb, false, false);

        // Rows 0..2 -> planes {0,1,2} (lanes 0-15) and {8,9,10} (lanes 16-31):
        // valid for both halves, store with full EXEC.
        #pragma unroll
        for (int r = 0; r < 3; ++r) {
            planes[so[r] + j0] = ca[r];
            planes[so[r] + j1] = cb[r];
        }
        // Rows 3..7 -> planes 3..7, only lanes 0-15 (single divergent region).
        if (half == 0) {
            #pragma unroll
            for (int r = 3; r < 8; ++r) {
                planes[so[r] + j0] = ca[r];
                planes[so[r] + j1] = cb[r];
            }
        }
    }
}

extern "C" void kernel_launch(void* const* d_in, const int* in_sizes, int n_in,
                              void* d_out, int out_size, void* d_ws, size_t ws_size,
                              hipStream_t stream) {
    (void)n_in; (void)out_size; (void)d_ws; (void)ws_size;

    const float* x    = (const float*)d_in[0];  // (N, C) f32
    const int*   erow = (const int*)  d_in[1];  // (E+N,) i32
    const int*   ecol = (const int*)  d_in[2];  // (E+N,) i32
    const float* eval = (const float*)d_in[3];  // (E+N,) f32
    float* out = (float*)d_out;                 // (11, N, C) f32

    const int NC   = in_sizes[0];               // N*C = 4,800,000 (divisible by 32)
    const int Etot = in_sizes[1];               // E + N = 1,700,000

    // plane 0 <- x
    hipMemcpyAsync(out, x, (size_t)NC * sizeof(float), hipMemcpyDeviceToDevice, stream);

    const int n4      = NC / 4;
    const int blkInit = (n4 + 255) / 256;
    const int blkScat = (Etot * 12 + 255) / 256;

    for (int m = 1; m <= KORD; ++m) {
        const float* prev = out + (size_t)(m - 1) * NC;
        float*       cur  = out + (size_t)m * NC;
        halve_kernel<<<blkInit, 256, 0, stream>>>(
            (const float4*)prev, (float4*)cur, n4);
        spmm_scatter_kernel<<<blkScat, 256, 0, stream>>>(
            prev, cur, erow, ecol, eval, Etot);
    }

    // Dense Bernstein combine, in place over all 11 planes.
    const int npairs = NC / 32;                 // 150,000 tile pairs (32 cols each)
    combine_wmma_kernel<<<1024, 256, 0, stream>>>(out, NC, npairs);
}